// TAGNet_74543452390037
// MI455X (gfx1250) — compile-verified
//
#include <hip/hip_runtime.h>
#include <hip/hip_bf16.h>

// ---------------------------------------------------------------------------
// TAGNet (3x TAGConv K=2, PReLU + exact-JAX threefry dropout) for gfx1250.
// Dense 64-wide GEMMs use V_WMMA_F32_16X16X4_F32 (fp32-exact WMMA).
// ---------------------------------------------------------------------------

#define FDIM 64

typedef float v2f __attribute__((ext_vector_type(2)));
typedef float v8f __attribute__((ext_vector_type(8)));

// ---------------- threefry2x32 (matches JAX) ------------------------------
__host__ __device__ static inline unsigned rotl32(unsigned x, int r) {
    return (x << r) | (x >> (32 - r));
}

__host__ __device__ static inline void threefry2x32(unsigned k0, unsigned k1,
                                                    unsigned x0, unsigned x1,
                                                    unsigned& o0, unsigned& o1) {
    const unsigned ks2 = k0 ^ k1 ^ 0x1BD11BDAu;
    x0 += k0; x1 += k1;
    // round block A (13,15,26,6)
    x0 += x1; x1 = rotl32(x1, 13); x1 ^= x0;
    x0 += x1; x1 = rotl32(x1, 15); x1 ^= x0;
    x0 += x1; x1 = rotl32(x1, 26); x1 ^= x0;
    x0 += x1; x1 = rotl32(x1, 6);  x1 ^= x0;
    x0 += k1; x1 += ks2 + 1u;
    // round block B (17,29,16,24)
    x0 += x1; x1 = rotl32(x1, 17); x1 ^= x0;
    x0 += x1; x1 = rotl32(x1, 29); x1 ^= x0;
    x0 += x1; x1 = rotl32(x1, 16); x1 ^= x0;
    x0 += x1; x1 = rotl32(x1, 24); x1 ^= x0;
    x0 += ks2; x1 += k0 + 2u;
    // A
    x0 += x1; x1 = rotl32(x1, 13); x1 ^= x0;
    x0 += x1; x1 = rotl32(x1, 15); x1 ^= x0;
    x0 += x1; x1 = rotl32(x1, 26); x1 ^= x0;
    x0 += x1; x1 = rotl32(x1, 6);  x1 ^= x0;
    x0 += k0; x1 += k1 + 3u;
    // B
    x0 += x1; x1 = rotl32(x1, 17); x1 ^= x0;
    x0 += x1; x1 = rotl32(x1, 29); x1 ^= x0;
    x0 += x1; x1 = rotl32(x1, 16); x1 ^= x0;
    x0 += x1; x1 = rotl32(x1, 24); x1 ^= x0;
    x0 += k1; x1 += ks2 + 4u;
    // A
    x0 += x1; x1 = rotl32(x1, 13); x1 ^= x0;
    x0 += x1; x1 = rotl32(x1, 15); x1 ^= x0;
    x0 += x1; x1 = rotl32(x1, 26); x1 ^= x0;
    x0 += x1; x1 = rotl32(x1, 6);  x1 ^= x0;
    x0 += ks2; x1 += k0 + 5u;
    o0 = x0; o1 = x1;
}

// ---------------- utility kernels -----------------------------------------
__global__ void tag_zero(float* __restrict__ p, long long n) {
    long long i = (long long)blockIdx.x * blockDim.x + threadIdx.x;
    if (i < n) p[i] = 0.0f;
}

__global__ void tag_degree(float* __restrict__ deg,
                           const long long* __restrict__ eidx,
                           const float* __restrict__ ew, int E) {
    int e = blockIdx.x * blockDim.x + threadIdx.x;
    if (e >= E) return;
    int c = (int)eidx[(size_t)E + e];
    atomicAdd(&deg[c], ew[e]);
}

__global__ void tag_dinv(float* __restrict__ d, int n) {
    int i = blockIdx.x * blockDim.x + threadIdx.x;
    if (i >= n) return;
    float v = d[i];
    d[i] = (v > 0.0f) ? rsqrtf(fmaxf(v, 1e-12f)) : 0.0f;
}

__global__ void tag_norm(float* __restrict__ norm,
                         const float* __restrict__ dinv,
                         const long long* __restrict__ eidx,
                         const float* __restrict__ ew, int E) {
    int e = blockIdx.x * blockDim.x + threadIdx.x;
    if (e >= E) return;
    int r = (int)eidx[e];
    int c = (int)eidx[(size_t)E + e];
    norm[e] = dinv[r] * ew[e] * dinv[c];
}

// One wave (32 lanes) per edge; each lane moves a float2 (2 of 64 features).
__global__ void tag_spmm(float* __restrict__ dst, const float* __restrict__ src,
                         const long long* __restrict__ eidx,
                         const float* __restrict__ norm, int E) {
    long long t = (long long)blockIdx.x * blockDim.x + threadIdx.x;
    int e = (int)(t >> 5);
    if (e >= E) return;
    int lane = (int)(t & 31);
    int r = (int)eidx[e];
    int c = (int)eidx[(size_t)E + e];
    float w = norm[e];
    float2 v = ((const float2*)(src + (size_t)r * FDIM))[lane];
    atomicAdd(&dst[(size_t)c * FDIM + 2 * lane],     v.x * w);
    atomicAdd(&dst[(size_t)c * FDIM + 2 * lane + 1], v.y * w);
}

// ---------------- WMMA GEMM: out[N,64] = A0@W[0] + A1@W[1] + A2@W[2] ------
// 4 waves/block; wave ct owns output tile [16 rows x cols ct*16..ct*16+15].
// K-loop: 3 sources x (64/4) steps of V_WMMA_F32_16X16X4_F32.
__global__ __launch_bounds__(128) void tag_gemm64(
    const float* __restrict__ A0, const float* __restrict__ A1,
    const float* __restrict__ A2, const float* __restrict__ W,
    float* __restrict__ out) {
    const int lane = threadIdx.x & 31;
    const int ct   = threadIdx.x >> 5;        // 0..3 column tile
    const int row0 = blockIdx.x * 16;
    const int m    = lane & 15;               // row (A) / col (B) within tile
    const int hoff = lane >> 4;               // 0: K={0,1}, 1: K={2,3}
    const int n    = ct * 16 + m;

    v8f c = {0.f, 0.f, 0.f, 0.f, 0.f, 0.f, 0.f, 0.f};
    const float* As[3] = {A0, A1, A2};
#pragma unroll
    for (int s = 0; s < 3; ++s) {
        const float*  A  = As[s];
        const float*  Wk = W + (size_t)s * FDIM * FDIM;
        const float2* ar = (const float2*)(A + (size_t)(row0 + m) * FDIM);
#pragma unroll
        for (int kk = 0; kk < FDIM; kk += 4) {
            float2 av = ar[(kk >> 1) + hoff];     // A[m][kk+2*hoff .. +1]
            int kb = kk + 2 * hoff;
            v2f a; a[0] = av.x; a[1] = av.y;
            v2f b; b[0] = Wk[(size_t)kb * FDIM + n];
                   b[1] = Wk[(size_t)(kb + 1) * FDIM + n];
            c = __builtin_amdgcn_wmma_f32_16x16x4_f32(
                    false, a, false, b, (short)0, c, false, false);
        }
    }
#pragma unroll
    for (int r = 0; r < 8; ++r)
        out[(size_t)(row0 + r + 8 * hoff) * FDIM + n] = c[r];
}

// ---------------- dropout (exact JAX threefry) + PReLU, in place ----------
__global__ void tag_dropout_prelu(float* __restrict__ h, long long n,
                                  unsigned k0, unsigned k1,
                                  const float* __restrict__ aptr) {
    long long i = (long long)blockIdx.x * blockDim.x + threadIdx.x;
    if (i >= n) return;
    unsigned half = (unsigned)(n >> 1);
    unsigned ui = (unsigned)i;
    unsigned x0 = (ui < half) ? ui : (ui - half);
    unsigned x1 = (ui < half) ? (ui + half) : ui;
    unsigned y0, y1;
    threefry2x32(k0, k1, x0, x1, y0, y1);
    unsigned bits = (ui < half) ? y0 : y1;
    float u = __uint_as_float((bits >> 9) | 0x3f800000u) - 1.0f;
    float v = h[i];
    v = (u < 0.5f) ? v * 2.0f : 0.0f;              // keep-prob 0.5, scale 1/0.5
    float a = *aptr;
    h[i] = (v >= 0.0f) ? v : a * v;
}

// ---------------- final layer: out[i] = sum_k h_k[i,:] . W2[k,:,0] --------
// one wave per node, shuffle reduction
__global__ void tag_final(const float* __restrict__ A0, const float* __restrict__ A1,
                          const float* __restrict__ A2, const float* __restrict__ W2,
                          float* __restrict__ out, int n) {
    long long t = (long long)blockIdx.x * blockDim.x + threadIdx.x;
    int node = (int)(t >> 5);
    int lane = (int)(t & 31);
    if (node >= n) return;
    size_t base = (size_t)node * FDIM;
    float acc = A0[base + lane] * W2[lane] + A0[base + lane + 32] * W2[lane + 32]
              + A1[base + lane] * W2[64 + lane] + A1[base + lane + 32] * W2[64 + lane + 32]
              + A2[base + lane] * W2[128 + lane] + A2[base + lane + 32] * W2[128 + lane + 32];
#pragma unroll
    for (int off = 16; off > 0; off >>= 1)
        acc += __shfl_down(acc, off, 32);
    if (lane == 0) out[node] = acc;
}

// ---------------------------------------------------------------------------
extern "C" void kernel_launch(void* const* d_in, const int* in_sizes, int n_in,
                              void* d_out, int out_size, void* d_ws, size_t ws_size,
                              hipStream_t stream) {
    const float*     x    = (const float*)d_in[0];
    const long long* eidx = (const long long*)d_in[1];
    const float*     ew   = (const float*)d_in[2];
    const float*     W0   = (const float*)d_in[3];
    const float*     W1   = (const float*)d_in[4];
    const float*     W2   = (const float*)d_in[5];
    const float*     a0   = (const float*)d_in[6];
    const float*     a1   = (const float*)d_in[7];
    float*           out  = (float*)d_out;

    const int N = in_sizes[0] / FDIM;   // 50000
    const int E = in_sizes[2];          // 800000
    const long long NF = (long long)N * FDIM;

    // workspace layout (floats)
    float* wsf  = (float*)d_ws;
    float* norm = wsf;                       // E
    float* dinv = norm + E;                  // N  (deg then dinv in place)
    float* h1   = dinv + N;                  // N*64
    float* h2   = h1 + NF;                   // N*64
    float* bufA = h2 + NF;                   // N*64
    float* bufB = bufA + NF;                 // N*64

    // host-side: dk = jax.random.split(jax.random.key(42), 2)
    unsigned ya0, ya1, yb0, yb1;
    threefry2x32(0u, 42u, 0u, 2u, ya0, ya1);   // pair (counts 0, 2)
    threefry2x32(0u, 42u, 1u, 3u, yb0, yb1);   // pair (counts 1, 3)
    const unsigned dk0_0 = ya0, dk0_1 = yb0;   // dk[0]
    const unsigned dk1_0 = ya1, dk1_1 = yb1;   // dk[1]

    const int TB = 256;
    dim3 blk(TB);
    const int gN   = (N + TB - 1) / TB;
    const int gE   = (E + TB - 1) / TB;
    const int gNF  = (int)((NF + TB - 1) / TB);
    const int gSp  = (int)(((long long)E * 32 + TB - 1) / TB);
    const int gFin = (int)(((long long)N * 32 + TB - 1) / TB);
    const int gGemm = N / 16;                  // N is a multiple of 16

    // normalization
    tag_zero<<<gN, blk, 0, stream>>>(dinv, N);
    tag_degree<<<gE, blk, 0, stream>>>(dinv, eidx, ew, E);
    tag_dinv<<<gN, blk, 0, stream>>>(dinv, N);
    tag_norm<<<gE, blk, 0, stream>>>(norm, dinv, eidx, ew, E);

    // ----- layer 0 -----
    tag_zero<<<gNF, blk, 0, stream>>>(h1, NF);
    tag_spmm<<<gSp, blk, 0, stream>>>(h1, x, eidx, norm, E);
    tag_zero<<<gNF, blk, 0, stream>>>(h2, NF);
    tag_spmm<<<gSp, blk, 0, stream>>>(h2, h1, eidx, norm, E);
    tag_gemm64<<<gGemm, dim3(128), 0, stream>>>(x, h1, h2, W0, bufA);
    tag_dropout_prelu<<<gNF, blk, 0, stream>>>(bufA, NF, dk0_0, dk0_1, a0);

    // ----- layer 1 -----
    tag_zero<<<gNF, blk, 0, stream>>>(h1, NF);
    tag_spmm<<<gSp, blk, 0, stream>>>(h1, bufA, eidx, norm, E);
    tag_zero<<<gNF, blk, 0, stream>>>(h2, NF);
    tag_spmm<<<gSp, blk, 0, stream>>>(h2, h1, eidx, norm, E);
    tag_gemm64<<<gGemm, dim3(128), 0, stream>>>(bufA, h1, h2, W1, bufB);
    tag_dropout_prelu<<<gNF, blk, 0, stream>>>(bufB, NF, dk1_0, dk1_1, a1);

    // ----- layer 2 (output dim 1) -----
    tag_zero<<<gNF, blk, 0, stream>>>(h1, NF);
    tag_spmm<<<gSp, blk, 0, stream>>>(h1, bufB, eidx, norm, E);
    tag_zero<<<gNF, blk, 0, stream>>>(h2, NF);
    tag_spmm<<<gSp, blk, 0, stream>>>(h2, h1, eidx, norm, E);
    tag_final<<<gFin, blk, 0, stream>>>(bufB, h1, h2, W2, out, N);
}